// Model_28681791602769
// MI455X (gfx1250) — compile-verified
//
#include <hip/hip_runtime.h>
#include <hip/hip_bf16.h>

// Split-KV (flash-decoding) stage-2 merge for MI455X (gfx1250, wave32).
// Memory-bound (~68 MB @ 23.3 TB/s ~= 3 us floor). The weighted reduction over
// the 16 KV-splits is folded into chained V_WMMA_F32_16X16X4_F32 ops:
//   D = A(16x4, w broadcast over rows) x B(4x16, att_out rows) + C
// 4 chained WMMAs cover K=16 splits; 8 column chunks cover D=128.

typedef __attribute__((ext_vector_type(2))) float v2f;
typedef __attribute__((ext_vector_type(8))) float v8f;

#define S_SPLITS 16
#define D_DIM    128
#define WAVES_PER_BLOCK 8

__global__ __launch_bounds__(256) void fd_merge_wmma(
    const float* __restrict__ att_out,      // [B,H,S,D]
    const float* __restrict__ att_lse,      // [B,H,S]
    const int*   __restrict__ kv_indptr,    // [B+1]
    const int*   __restrict__ num_kv_splits,// [B]
    float*       __restrict__ out,          // [B,H,D]
    int BH, int H)
{
    const int lane = threadIdx.x & 31;
    const int wave = threadIdx.x >> 5;
    const int bh   = blockIdx.x * WAVES_PER_BLOCK + wave;
    if (bh >= BH) return;              // wave-uniform: EXEC stays all-1 for live waves
    const int b = bh / H;

    // ---- per-batch split validity + softmax weights over LSE ----------------
    const int seq_len = kv_indptr[b + 1] - kv_indptr[b];
    const int splits  = num_kv_splits[b];
    // cdiv(cdiv(seq_len, splits), 32) * 32   (seq_len >= 32, splits >= 1)
    const int per_split = ((((seq_len + splits - 1) / splits) + 31) / 32) * 32;

    const float* lse = att_lse + (size_t)bh * S_SPLITS;
    float lv[S_SPLITS];
    bool  valid[S_SPLITS];
    float m = -3.402823466e+38f;       // split 0 is always valid -> m finite
    #pragma unroll
    for (int s = 0; s < S_SPLITS; ++s) {
        valid[s] = (per_split * s) < seq_len;
        lv[s] = lse[s];
        if (valid[s]) m = fmaxf(m, lv[s]);
    }
    float w[S_SPLITS];
    float esum = 0.f;
    #pragma unroll
    for (int s = 0; s < S_SPLITS; ++s) {
        float e = valid[s] ? expf(lv[s] - m) : 0.f;
        w[s] = e;
        esum += e;
    }
    const float r = 1.0f / esum;       // fold normalization into the weights
    #pragma unroll
    for (int s = 0; s < S_SPLITS; ++s) w[s] *= r;

    // ---- WMMA operand staging ----------------------------------------------
    // A (16x4 f32): lane m = lane&15; VGPR0 holds K = (lo half ? 4c+0 : 4c+2),
    // VGPR1 holds K = (lo ? 4c+1 : 4c+3). A rows are all w[k] (broadcast).
    // B (4x16 f32): same k-per-slot convention; row k striped over 16 lanes.
    const bool hi  = lane >= 16;
    const int  col = lane & 15;
    const int  so0 = hi ? 2 : 0;
    const int  so1 = hi ? 3 : 1;

    float wa0[4], wa1[4];
    #pragma unroll
    for (int c = 0; c < 4; ++c) {
        wa0[c] = hi ? w[4 * c + 2] : w[4 * c + 0];
        wa1[c] = hi ? w[4 * c + 3] : w[4 * c + 1];
    }

    const float* base = att_out + (size_t)bh * (S_SPLITS * D_DIM);
    float*       outp = out + (size_t)bh * D_DIM;

    // 8 chunks of 16 d-values, processed in pairs -> full 32-lane stores.
    #pragma unroll
    for (int p = 0; p < 4; ++p) {
        const int d0 = (2 * p) * 16 + col;
        const int d1 = (2 * p + 1) * 16 + col;
        v8f acc0 = {0.f, 0.f, 0.f, 0.f, 0.f, 0.f, 0.f, 0.f};
        v8f acc1 = {0.f, 0.f, 0.f, 0.f, 0.f, 0.f, 0.f, 0.f};
        #pragma unroll
        for (int c = 0; c < 4; ++c) {
            v2f A;
            A.x = wa0[c];
            A.y = wa1[c];
            const float* r0 = base + (size_t)(4 * c + so0) * D_DIM;
            const float* r1 = base + (size_t)(4 * c + so1) * D_DIM;
            v2f B0, B1;
            // att_out is streamed exactly once -> non-temporal loads
            B0.x = __builtin_nontemporal_load(r0 + d0);
            B0.y = __builtin_nontemporal_load(r1 + d0);
            B1.x = __builtin_nontemporal_load(r0 + d1);
            B1.y = __builtin_nontemporal_load(r1 + d1);
            acc0 = __builtin_amdgcn_wmma_f32_16x16x4_f32(
                       false, A, false, B0, (short)0, acc0, false, false);
            acc1 = __builtin_amdgcn_wmma_f32_16x16x4_f32(
                       false, A, false, B1, (short)0, acc1, false, false);
        }
        // All D rows identical (A rows identical): acc[0] lane l = o[chunk*16 + (l&15)].
        // Pair the two chunks into one fully-coalesced 32-lane store.
        float v = hi ? acc1[0] : acc0[0];
        __builtin_nontemporal_store(v, outp + p * 32 + lane);
    }
}

extern "C" void kernel_launch(void* const* d_in, const int* in_sizes, int n_in,
                              void* d_out, int out_size, void* d_ws, size_t ws_size,
                              hipStream_t stream) {
    const float* att_out       = (const float*)d_in[0];
    const float* att_lse       = (const float*)d_in[1];
    // d_in[2] = q, d_in[3] = v_buffer : shape-only inputs, unused by the math
    const int*   kv_indptr     = (const int*)d_in[4];
    const int*   num_kv_splits = (const int*)d_in[5];
    float*       out           = (float*)d_out;

    const int BH = in_sizes[1] / S_SPLITS;   // att_lse = [B,H,S]
    const int Bn = in_sizes[4] - 1;          // kv_indptr = [B+1]
    const int Hn = BH / Bn;

    const int blocks = (BH + WAVES_PER_BLOCK - 1) / WAVES_PER_BLOCK;
    fd_merge_wmma<<<blocks, 32 * WAVES_PER_BLOCK, 0, stream>>>(
        att_out, att_lse, kv_indptr, num_kv_splits, out, BH, Hn);
}